// ProjectLayer_35210141893407
// MI455X (gfx1250) — compile-verified
//
#include <hip/hip_runtime.h>
#include <cstdint>

// ---- problem constants (from reference) ----
#define HMW 240
#define HMH 128
#define NB  128000            // 80*80*20
#define MBOX 10
#define IMG_W 960.0f
#define IMG_H 512.0f

#ifndef __has_builtin
#define __has_builtin(x) 0
#endif

typedef unsigned u32x4 __attribute__((ext_vector_type(4)));
typedef int      i32x4 __attribute__((ext_vector_type(4)));
typedef int      i32x8 __attribute__((ext_vector_type(8)));
typedef float    f32x4 __attribute__((ext_vector_type(4)));

static __device__ __forceinline__ float clampf(float v, float lo, float hi) {
    return fminf(fmaxf(v, lo), hi);
}

__global__ __launch_bounds__(1024)
void project_layer_kernel(
    const float* __restrict__ hm,      // [B,J,HMH,HMW]
    const float* __restrict__ gc,      // [B,3]
    const float* __restrict__ center,  // [B,2]
    const float* __restrict__ scale,   // [B]
    const float* __restrict__ Rm,      // [B,3,3]
    const float* __restrict__ Tm,      // [B,3,1]
    const float* __restrict__ fl,      // [B,2]
    const float* __restrict__ cp,      // [B,2]
    const float* __restrict__ kd,      // [B,3]
    const float* __restrict__ pd,      // [B,2]
    const float* __restrict__ bbx,     // [B,M,4]
    const float* __restrict__ dep,     // [B,M]
    const float* __restrict__ dnf,     // [B]
    int J,
    float* __restrict__ out_cubes,     // [B,J,NB]
    float* __restrict__ out_grids)     // [B,NB,3]
{
    extern __shared__ float smem[];    // HMW*HMH floats = 122,880 bytes
    const int blk = blockIdx.x;
    const int b  = blk / J;
    const int j  = blk - b * J;
    const int tid = threadIdx.x;
    const int nthreads = blockDim.x;

    // ---------------- stage heatmap[b, j] into LDS ----------------
    const float* src = hm + ((size_t)(b * J + j)) * (size_t)(HMW * HMH);

#if __has_builtin(__builtin_amdgcn_tensor_load_to_lds)
    // CDNA5 Tensor Data Mover: one DMA moves the whole 240x128 f32 tile.
    if (tid < 32) {                    // single wave issues the TDM op
        uint64_t ga = (uint64_t)(uintptr_t)src;
        // D# group 0: count=1, lds_addr=0, global_addr, type=2 ("image")
        u32x4 g0;
        g0.x = 1u;                                  // count=1, user mode
        g0.y = 0u;                                  // lds_addr (bytes)
        g0.z = (unsigned)(ga & 0xffffffffu);        // global_addr[31:0]
        g0.w = (unsigned)((ga >> 32) & 0x1ffffffu)  // global_addr[56:32]
               | (2u << 30);                        // type=2
        // D# group 1: data_size=4B, dims/strides for a 240x128 2D tile
        i32x8 g1;
        g1[0] = (int)(2u << 16);                    // data_size=2 (4 bytes)
        g1[1] = (int)((unsigned)HMW << 16);         // tensor_dim0[15:0] in [31:16]
        g1[2] = (int)((unsigned)HMH << 16);         // tensor_dim0 hi=0 | tensor_dim1 lo
        g1[3] = (int)((unsigned)HMW << 16);         // tensor_dim1 hi=0 | tile_dim0=240
        g1[4] = (int)HMH;                           // tile_dim1=128, tile_dim2=0
        g1[5] = (int)HMW;                           // tensor_dim0_stride low 32 = 240
        g1[6] = 0;                                  // stride0 hi | tensor_dim1_stride lo
        g1[7] = 0;                                  // tensor_dim1_stride hi
        i32x4 gz = {0, 0, 0, 0};                    // groups 2/3 unused (2D tensor)
#if __clang_major__ >= 23
        i32x8 gz8 = {0, 0, 0, 0, 0, 0, 0, 0};
        __builtin_amdgcn_tensor_load_to_lds(g0, g1, gz, gz, gz8, 0);
#else
        __builtin_amdgcn_tensor_load_to_lds(g0, g1, gz, gz, 0);
#endif
#if __has_builtin(__builtin_amdgcn_s_wait_tensorcnt)
        __builtin_amdgcn_s_wait_tensorcnt(0);
#else
        asm volatile("s_wait_tensorcnt 0" ::: "memory");
#endif
    }
    __syncthreads();
#else
    // Fallback: per-lane async global->LDS b128 copies (ASYNCcnt path)
    {
        const unsigned ldsbase = (unsigned)(uintptr_t)smem;
        const int nchunks = (HMW * HMH) / 4;       // 7680 b128 chunks
        for (int i = tid; i < nchunks; i += nthreads) {
            unsigned dst = ldsbase + (unsigned)i * 16u;
            uint64_t ga  = (uint64_t)(uintptr_t)(src + (size_t)i * 4);
            asm volatile("global_load_async_to_lds_b128 %0, %1, off"
                         :: "v"(dst), "v"(ga) : "memory");
        }
        asm volatile("s_wait_asynccnt 0" ::: "memory");
        __syncthreads();
    }
#endif

    // ---------------- per-sample (uniform) parameters ----------------
    const float gc0 = gc[b*3+0], gc1 = gc[b*3+1], gc2 = gc[b*3+2];
    const float c0  = center[b*2+0], c1 = center[b*2+1];
    const float sc  = scale[b];
    const float R00 = Rm[b*9+0], R01 = Rm[b*9+1], R02 = Rm[b*9+2];
    const float R10 = Rm[b*9+3], R11 = Rm[b*9+4], R12 = Rm[b*9+5];
    const float R20 = Rm[b*9+6], R21 = Rm[b*9+7], R22 = Rm[b*9+8];
    const float T0  = Tm[b*3+0], T1 = Tm[b*3+1], T2 = Tm[b*3+2];
    const float f0  = fl[b*2+0], f1 = fl[b*2+1];
    const float cc0 = cp[b*2+0], cc1 = cp[b*2+1];
    const float k0  = kd[b*3+0], k1 = kd[b*3+1], k2 = kd[b*3+2];
    const float p0  = pd[b*2+0], p1 = pd[b*2+1];
    const float dn  = dnf[b];

    const float width  = c0 * 2.0f;
    const float height = c1 * 2.0f;
    const float mx     = fmaxf(width, height);
    const float hh     = 200.0f * sc;
    const float a0     = IMG_W / hh;
    const float a1     = IMG_H / hh;
    const float t0     = IMG_W * (-c0 / hh + 0.5f);
    const float t1     = IMG_H * (-c1 / hh + 0.5f);
    const float sig2inv = 1.0f / (2.0f * 200.0f * 200.0f);

    float bx1[MBOX], by1[MBOX], bx2[MBOX], by2[MBOX], dm[MBOX], vl[MBOX];
#pragma unroll
    for (int m = 0; m < MBOX; ++m) {
        bx1[m] = bbx[(b*MBOX + m)*4 + 0];
        by1[m] = bbx[(b*MBOX + m)*4 + 1];
        bx2[m] = bbx[(b*MBOX + m)*4 + 2];
        by2[m] = bbx[(b*MBOX + m)*4 + 3];
        float d = dep[b*MBOX + m];
        dm[m] = d * dn;
        vl[m] = (d >= 0.0f) ? 1.0f : 0.0f;
    }

    float* cub = out_cubes + ((size_t)(b * J + j)) * (size_t)NB;

    // ---------------- per-voxel loop: 4 consecutive z-bins per thread ----------------
    const int NQ = NB / 4;                         // 32000 quads; 20 % 4 == 0 so a
    for (int q = tid; q < NQ; q += nthreads) {     // quad never crosses a y-column
        const int n  = q * 4;
        const int xi = n / 1600;                   // 80*20
        const int rem = n - xi * 1600;
        const int yi = rem / 20;
        const int zi = rem - yi * 20;

        const float px = -4000.0f + (float)xi * (8000.0f / 79.0f) + gc0;
        const float py = -4000.0f + (float)yi * (8000.0f / 79.0f) + gc1;
        const float pz0 = -1000.0f + (float)zi * (2000.0f / 19.0f) + gc2;

        if (j == 0) {
            // 12 contiguous floats: {px,py,pz_r} for r=0..3 -> three b128 stores
            const float pza = pz0;
            const float pzb = pz0 + (2000.0f / 19.0f);
            const float pzc = pz0 + 2.0f * (2000.0f / 19.0f);
            const float pzd = pz0 + 3.0f * (2000.0f / 19.0f);
            f32x4* gp = (f32x4*)(out_grids + ((size_t)b * NB + (size_t)n) * 3);
            gp[0] = (f32x4){px, py, pza, px};
            gp[1] = (f32x4){py, pzb, px, py};
            gp[2] = (f32x4){pzc, px, py, pzd};
        }

        // shared part of camera transform (only pz varies within the quad)
        const float xb = R00*px + R01*py - T0;
        const float yb = R10*px + R11*py - T1;
        const float zb = R20*px + R21*py - T2;

        f32x4 res;
#pragma unroll
        for (int r = 0; r < 4; ++r) {
            const float pz = pz0 + (float)r * (2000.0f / 19.0f);
            const float xc = xb + R02 * pz;
            const float yc = yb + R12 * pz;
            const float zc = zb + R22 * pz;
            const float zd = zc + 1e-5f;
            const float y0 = xc / zd;
            const float y1 = yc / zd;
            const float r2 = y0*y0 + y1*y1;
            const float radial = 1.0f + r2*(k0 + r2*(k1 + r2*k2));
            const float tanv = p0*y1 + p1*y0;
            const float rt = radial + 2.0f*tanv;
            float xp = f0*(y0*rt + p1*r2) + cc0;
            float yp = f1*(y1*rt + p0*r2) + cc1;

            const bool bound = (xp >= 0.0f) & (yp >= 0.0f) & (xp < width) & (yp < height);

            xp = clampf(xp, -1.0f, mx);
            yp = clampf(yp, -1.0f, mx);

            // affine to image coords, then to heatmap coords
            const float hx = (a0*xp + t0) * ((float)HMW / IMG_W);
            const float hy = (a1*yp + t1) * ((float)HMH / IMG_H);

            // normalized sample coords, clipped to [-1.1, 1.1]
            const float sgx = clampf(hx * (2.0f / (HMW - 1.0f)) - 1.0f, -1.1f, 1.1f);
            const float sgy = clampf(hy * (2.0f / (HMH - 1.0f)) - 1.0f, -1.1f, 1.1f);
            const float sx = (sgx + 1.0f) * 0.5f * (HMW - 1.0f);
            const float sy = (sgy + 1.0f) * 0.5f * (HMH - 1.0f);

            // bilinear from LDS (zeros padding)
            const float x0f = floorf(sx), y0f = floorf(sy);
            const float wx = sx - x0f,   wy = sy - y0f;
            const int ix0 = (int)x0f,    iy0 = (int)y0f;
            const int ix1 = ix0 + 1,     iy1 = iy0 + 1;

            const bool okx0 = (ix0 >= 0) & (ix0 <= HMW - 1);
            const bool okx1 = (ix1 >= 0) & (ix1 <= HMW - 1);
            const bool oky0 = (iy0 >= 0) & (iy0 <= HMH - 1);
            const bool oky1 = (iy1 >= 0) & (iy1 <= HMH - 1);

            const int cx0 = min(max(ix0, 0), HMW - 1);
            const int cx1 = min(max(ix1, 0), HMW - 1);
            const int cy0 = min(max(iy0, 0), HMH - 1);
            const int cy1 = min(max(iy1, 0), HMH - 1);

            const float v00 = smem[cy0 * HMW + cx0];
            const float v01 = smem[cy0 * HMW + cx1];
            const float v10 = smem[cy1 * HMW + cx0];
            const float v11 = smem[cy1 * HMW + cx1];

            const float w00 = (okx0 & oky0) ? (1.0f - wx) * (1.0f - wy) : 0.0f;
            const float w01 = (okx1 & oky0) ? wx * (1.0f - wy)          : 0.0f;
            const float w10 = (okx0 & oky1) ? (1.0f - wx) * wy          : 0.0f;
            const float w11 = (okx1 & oky1) ? wx * wy                   : 0.0f;

            const float sval = v00*w00 + v01*w01 + v10*w10 + v11*w11;

            // max over bboxes of inbox * gaussian(depth) * valid
            float maxw = 0.0f;
#pragma unroll
            for (int m = 0; m < MBOX; ++m) {
                const bool inb = bound & (hx >= bx1[m]) & (hy >= by1[m])
                                       & (hx <= bx2[m]) & (hy <= by2[m]);
                const float d = zc - dm[m];
                const float g = __expf(-(d * d) * sig2inv) * vl[m];
                maxw = fmaxf(maxw, inb ? g : 0.0f);
            }

            const float v = maxw * sval;
            // nan_to_num + clip(0,1): fmaxf/fminf drop NaN operands
            res[r] = fminf(fmaxf(v, 0.0f), 1.0f);
        }
        *(f32x4*)(cub + n) = res;                  // one b128 store per quad
    }
}

extern "C" void kernel_launch(void* const* d_in, const int* in_sizes, int n_in,
                              void* d_out, int out_size, void* d_ws, size_t ws_size,
                              hipStream_t stream) {
    const float* hm     = (const float*)d_in[0];
    const float* gc     = (const float*)d_in[1];
    const float* center = (const float*)d_in[2];
    const float* scale  = (const float*)d_in[3];
    const float* Rm     = (const float*)d_in[4];
    const float* Tm     = (const float*)d_in[5];
    const float* fl     = (const float*)d_in[6];
    const float* cp     = (const float*)d_in[7];
    const float* kd     = (const float*)d_in[8];
    const float* pd     = (const float*)d_in[9];
    const float* bbx    = (const float*)d_in[10];
    const float* dep    = (const float*)d_in[11];
    const float* dnf    = (const float*)d_in[12];

    const int B = in_sizes[3];                         // scale is [B]
    const int J = in_sizes[0] / (B * HMH * HMW);       // heatmaps [B,J,H,W]

    float* cubes = (float*)d_out;
    float* grids = cubes + (size_t)B * (size_t)J * (size_t)NB;

    dim3 grid(B * J);
    dim3 block(1024);
    size_t shmem = (size_t)HMW * HMH * sizeof(float);  // 122,880 B (fits CDNA5 320KB LDS)

    hipLaunchKernelGGL(project_layer_kernel, grid, block, shmem, stream,
                       hm, gc, center, scale, Rm, Tm, fl, cp, kd, pd, bbx, dep, dnf,
                       J, cubes, grids);
}